// LightEncoder_27195732918763
// MI455X (gfx1250) — compile-verified
//
#include <hip/hip_runtime.h>
#include <hip/hip_bf16.h>

typedef float v2f __attribute__((ext_vector_type(2)));
typedef float v8f __attribute__((ext_vector_type(8)));

#define NEG_BIG (-1.0e9f)

static __device__ __forceinline__ v8f wmma4(v2f a, v2f b, v8f c) {
  // D = A(16x4 f32) x B(4x16 f32) + C(16x16 f32)  -> v_wmma_f32_16x16x4_f32
  return __builtin_amdgcn_wmma_f32_16x16x4_f32(false, a, false, b, (short)0, c,
                                               false, false);
}

// gfx1250 async LDS-direct copy: 16 bytes global -> LDS per lane, ASYNCcnt.
static __device__ __forceinline__ void async_ld_b128(unsigned lds_off,
                                                     const float* gptr) {
  asm volatile("global_load_async_to_lds_b128 %0, %1, off"
               :: "v"(lds_off), "v"(gptr)
               : "memory");
}
static __device__ __forceinline__ void wait_asynccnt0() {
  asm volatile("s_wait_asynccnt 0x0" ::: "memory");
}
static __device__ __forceinline__ unsigned lds_off_of(const void* p) {
  // generic shared-aperture pointer: addr[31:0] == LDS byte offset (ISA 10.2)
  return (unsigned)(uintptr_t)p;
}

// ---------------------------------------------------------------------------
// WMMA GEMM:  Y[M,N] = W[M,K] * X[K,N] + bias[M] (+ res[M,N]) (ReLU?)
// 4-wave workgroup -> 64(M) x 64(N) tile; wave w owns rows [m0+16w, m0+16w+16).
// B panel (KB=32 x 64) staged in LDS via global_load_async_to_lds_b128,
// double-buffered (16 KB), drained with s_wait_asynccnt + barrier.
// A layout: lane l -> M = l%16, K pair {2*(l/16), 2*(l/16)+1} per 4-chunk
// B layout: lane l -> N = l%16, same K split across lane halves
// C layout: lane l -> N = l%16, VGPR r -> M = r + 8*(l/16)
// ---------------------------------------------------------------------------
#define KB 32

__global__ __launch_bounds__(128) void gemm16(
    const float* __restrict__ W, const float* __restrict__ X,
    const float* __restrict__ bias, const float* __restrict__ res,
    float* __restrict__ Y, int M, int N, int K, int do_relu) {
  __shared__ float smem[2][KB * 64];

  const int tid = threadIdx.x;
  const int wave = tid >> 5;
  const int lane = tid & 31;
  const int hi = lane >> 4;  // 0 / 1 lane half
  const int lm = lane & 15;
  const int m0 = blockIdx.y * 64 + wave * 16;
  const int n0 = blockIdx.x * 64;

  v8f acc0 = {}, acc1 = {}, acc2 = {}, acc3 = {};
  const float* wrow = W + (size_t)(m0 + lm) * K;

  // cooperative async stage of chunk k0 into smem[buf]
  auto stage = [&](int buf, int k0) {
#pragma unroll
    for (int i = 0; i < 4; ++i) {
      const int flat = i * 128 + tid;       // 512 x b128 = 8 KB
      const int kr = flat >> 4;             // 0..31
      const int nn = (flat & 15) * 4;       // 0..60
      async_ld_b128(lds_off_of(&smem[buf][flat * 4]),
                    X + (size_t)(k0 + kr) * N + n0 + nn);
    }
  };

  stage(0, 0);
  int cur = 0;
  for (int k0 = 0; k0 < K; k0 += KB) {
    wait_asynccnt0();     // my quarter of smem[cur] landed
    __syncthreads();      // everyone's quarter landed; prev reads of nxt done
    if (k0 + KB < K) {
      stage(cur ^ 1, k0 + KB);                         // overlap next panel
      __builtin_prefetch(wrow + k0 + KB, 0, 1);        // next A-row chunk
    }
    const float* bp = smem[cur];
#pragma unroll
    for (int kk = 0; kk < KB; kk += 4) {
      const int ka = kk + 2 * hi;
      v2f a;
      a.x = wrow[k0 + ka];
      a.y = wrow[k0 + ka + 1];
      const float* br0 = bp + ka * 64;
      const float* br1 = br0 + 64;
      v2f b0, b1, b2, b3;
      b0.x = br0[lm];      b0.y = br1[lm];
      b1.x = br0[16 + lm]; b1.y = br1[16 + lm];
      b2.x = br0[32 + lm]; b2.y = br1[32 + lm];
      b3.x = br0[48 + lm]; b3.y = br1[48 + lm];
      acc0 = wmma4(a, b0, acc0);
      acc1 = wmma4(a, b1, acc1);
      acc2 = wmma4(a, b2, acc2);
      acc3 = wmma4(a, b3, acc3);
    }
    cur ^= 1;
  }

#pragma unroll
  for (int r = 0; r < 8; ++r) {
    const int m = m0 + r + 8 * hi;
    const float bv = bias[m];
    float v0 = acc0[r] + bv, v1 = acc1[r] + bv, v2 = acc2[r] + bv,
          v3 = acc3[r] + bv;
    if (res) {
      const float* rr = res + (size_t)m * N + n0;
      v0 += rr[lm]; v1 += rr[16 + lm]; v2 += rr[32 + lm]; v3 += rr[48 + lm];
    }
    if (do_relu) {
      v0 = fmaxf(v0, 0.f); v1 = fmaxf(v1, 0.f);
      v2 = fmaxf(v2, 0.f); v3 = fmaxf(v3, 0.f);
    }
    float* yr = Y + (size_t)m * N + n0;
    yr[lm] = v0; yr[16 + lm] = v1; yr[32 + lm] = v2; yr[48 + lm] = v3;
  }
}

// ---------------------------------------------------------------------------
// Windowed multi-head attention (d = 64, HALF_WIN = 8), flash-style.
// qkv: [3C, N] rows (Q rows [0,C), K rows [C,2C), V rows [2C,3C)), N = H*W.
// One wave per (head, 16-query tile).  Only key tiles intersecting the ±8
// window are visited; exp(-1e9 - m) == 0 in fp32, so this is exact.
//   Scores:   S^T = Kmat(16keys x 64) * Q^T(64 x 16q)   [16 WMMA / key tile]
//   Softmax:  lane == query (C-layout N dim), stats via 1x shfl_xor(16)
//   Output:   O^T += V^T(64 x 16keys) * P^T(16keys x 16q) [16 WMMA / key tile]
// out: attention result in [C, N] layout.
// ---------------------------------------------------------------------------
__global__ void attn_win(const float* __restrict__ qkv, float* __restrict__ out,
                         int C, int H, int W) {
  const int N = H * W;
  const int lane = threadIdx.x & 31;
  const int hi = lane >> 4;
  const int lm = lane & 15;
  const int head = blockIdx.y;
  const int qbase = blockIdx.x * 16;

  const float* Qp = qkv + (size_t)(head * 64) * N;
  const float* Kp = qkv + (size_t)(C + head * 64) * N;
  const float* Vp = qkv + (size_t)(2 * C + head * 64) * N;

  // Preload Q^T (scaled by 1/sqrt(64)) for this query tile: 16 K-chunks of 4.
  v2f qb[16];
#pragma unroll
  for (int c = 0; c < 16; ++c) {
    const int dd = 4 * c + 2 * hi;
    qb[c].x = 0.125f * Qp[(size_t)dd * N + qbase + lm];
    qb[c].y = 0.125f * Qp[(size_t)(dd + 1) * N + qbase + lm];
  }

  const int hq = qbase / W;           // uniform across tile (W % 16 == 0)
  const int wq = (qbase % W) + lm;    // per-lane query column
  const int qct = (qbase % W) / 16;   // query column tile

  v8f o0 = {}, o1 = {}, o2 = {}, o3 = {};
  float m_run = NEG_BIG;
  float s_run = 0.f;

  const int hk0 = (hq - 8 > 0) ? hq - 8 : 0;
  const int hk1 = (hq + 8 < H - 1) ? hq + 8 : H - 1;
  const int wt0 = (qct - 1 > 0) ? qct - 1 : 0;
  const int wt1 = (qct + 1 < W / 16 - 1) ? qct + 1 : W / 16 - 1;

  for (int hk = hk0; hk <= hk1; ++hk) {
    for (int wt = wt0; wt <= wt1; ++wt) {
      const int kbase = hk * W + wt * 16;

      // ---- S^T = K * Q^T --------------------------------------------------
      v8f st = {};
#pragma unroll
      for (int c = 0; c < 16; ++c) {
        const int dd = 4 * c + 2 * hi;
        v2f a;
        a.x = Kp[(size_t)dd * N + kbase + lm];
        a.y = Kp[(size_t)(dd + 1) * N + kbase + lm];
        st = wmma4(a, qb[c], st);
      }

      // ---- window mask + tile max (lane == query) -------------------------
      float p[8];
      float tmax = NEG_BIG;
#pragma unroll
      for (int r = 0; r < 8; ++r) {
        const int kw = wt * 16 + r + 8 * hi;  // key column (row diff already ok)
        const int dw = wq - kw;
        float s = st[r];
        s = (dw >= -8 && dw <= 8) ? s : NEG_BIG;
        p[r] = s;
        tmax = fmaxf(tmax, s);
      }
      tmax = fmaxf(tmax, __shfl_xor(tmax, 16, 32));

      const float mnew = fmaxf(m_run, tmax);
      const float rescale = __expf(m_run - mnew);
      float tsum = 0.f;
#pragma unroll
      for (int r = 0; r < 8; ++r) {
        p[r] = __expf(p[r] - mnew);
        tsum += p[r];
      }
      tsum += __shfl_xor(tsum, 16, 32);
      s_run = s_run * rescale + tsum;
      m_run = mnew;

      // rescale accumulators (lane == query in O^T C-layout)
#pragma unroll
      for (int r = 0; r < 8; ++r) {
        o0[r] *= rescale; o1[r] *= rescale;
        o2[r] *= rescale; o3[r] *= rescale;
      }

      // ---- build P^T B-operand chunks (half-wave swap + select) -----------
      float sp[8];
#pragma unroll
      for (int r = 0; r < 8; ++r) sp[r] = __shfl_xor(p[r], 16, 32);
      v2f pb0, pb1, pb2, pb3;
      pb0.x = hi ? sp[2] : p[0];  pb0.y = hi ? sp[3] : p[1];
      pb1.x = hi ? sp[6] : p[4];  pb1.y = hi ? sp[7] : p[5];
      pb2.x = hi ? p[2] : sp[0];  pb2.y = hi ? p[3] : sp[1];
      pb3.x = hi ? p[6] : sp[4];  pb3.y = hi ? p[7] : sp[5];

      // ---- O^T += V^T * P^T ----------------------------------------------
#pragma unroll
      for (int t = 0; t < 4; ++t) {
        const float* vrow = Vp + (size_t)(16 * t + lm) * N + kbase + 2 * hi;
        v2f a0, a1, a2, a3;
        a0.x = vrow[0];  a0.y = vrow[1];
        a1.x = vrow[4];  a1.y = vrow[5];
        a2.x = vrow[8];  a2.y = vrow[9];
        a3.x = vrow[12]; a3.y = vrow[13];
        v8f acc = (t == 0) ? o0 : (t == 1) ? o1 : (t == 2) ? o2 : o3;
        acc = wmma4(a0, pb0, acc);
        acc = wmma4(a1, pb1, acc);
        acc = wmma4(a2, pb2, acc);
        acc = wmma4(a3, pb3, acc);
        if (t == 0) o0 = acc; else if (t == 1) o1 = acc;
        else if (t == 2) o2 = acc; else o3 = acc;
      }
    }
  }

  const float inv = 1.f / s_run;
#pragma unroll
  for (int t = 0; t < 4; ++t) {
    const v8f acc = (t == 0) ? o0 : (t == 1) ? o1 : (t == 2) ? o2 : o3;
#pragma unroll
    for (int r = 0; r < 8; ++r) {
      const int dd = 16 * t + r + 8 * hi;
      out[(size_t)(head * 64 + dd) * N + qbase + lm] = acc[r] * inv;
    }
  }
}

// ---------------------------------------------------------------------------
// Direct 3x3 stride-2 pad-1 conv (small: ~1.3 GMAC total across stages)
// ---------------------------------------------------------------------------
__global__ void conv3x3_s2(const float* __restrict__ X,
                           const float* __restrict__ Wt,
                           const float* __restrict__ b, float* __restrict__ Y,
                           int Cin, int Hin, int Win, int Cout) {
  const int Hout = Hin / 2, Wout = Win / 2;
  const int total = Cout * Hout * Wout;
  const int idx = blockIdx.x * blockDim.x + threadIdx.x;
  if (idx >= total) return;
  const int wo = idx % Wout;
  const int ho = (idx / Wout) % Hout;
  const int co = idx / (Wout * Hout);
  float acc = b[co];
  for (int ci = 0; ci < Cin; ++ci) {
    const float* xp = X + (size_t)ci * Hin * Win;
    const float* wp = Wt + (size_t)(co * Cin + ci) * 9;
#pragma unroll
    for (int kh = 0; kh < 3; ++kh) {
      const int ih = ho * 2 + kh - 1;
      if (ih < 0 || ih >= Hin) continue;
#pragma unroll
      for (int kw = 0; kw < 3; ++kw) {
        const int iw = wo * 2 + kw - 1;
        if (iw < 0 || iw >= Win) continue;
        acc = fmaf(xp[ih * Win + iw], wp[kh * 3 + kw], acc);
      }
    }
  }
  Y[idx] = acc;
}

// ---------------------------------------------------------------------------
extern "C" void kernel_launch(void* const* d_in, const int* in_sizes, int n_in,
                              void* d_out, int out_size, void* d_ws,
                              size_t ws_size, hipStream_t stream) {
  (void)in_sizes; (void)n_in; (void)out_size; (void)ws_size;

  const float* x      = (const float*)d_in[0];  // (1,512,64,64)
  const float* init_w = (const float*)d_in[1];  // (512,512,1,1)
  const float* init_b = (const float*)d_in[2];  // (512,)
  // per-layer params at 3 + 8*i: in_w,in_b,out_w,out_b,proj_w,proj_b,down_w,down_b

  float* out = (float*)d_out;
  // output layout: h(128*8*8) | feat0(512*64*64) | feat1(256*32*32) | feat2(128*16*16)
  float* feat0 = out + 8192;
  float* feat1 = feat0 + 512 * 4096;
  float* feat2 = feat1 + 256 * 1024;

  float* ws    = (float*)d_ws;
  float* hbuf  = ws;                       //  512*4096
  float* qkvb  = hbuf + 512 * 4096;        // 1536*4096
  float* attnb = qkvb + 1536 * 4096;       //  512*4096
  float* ybuf  = attnb + 512 * 4096;       //  512*4096

  const int Cs[3]    = {512, 256, 128};
  const int Hs[3]    = {64, 32, 16};
  const int Heads[3] = {8, 4, 2};
  const int Dn[3]    = {256, 128, 128};
  float* feats[3]    = {feat0, feat1, feat2};

  // init 1x1 conv + relu: h = relu(init_w * x + init_b), [512, 4096]
  gemm16<<<dim3(4096 / 64, 512 / 64), 128, 0, stream>>>(
      init_w, x, init_b, nullptr, hbuf, 512, 4096, 512, 1);

  const float* cur = hbuf;
  for (int i = 0; i < 3; ++i) {
    const int C = Cs[i], H = Hs[i], W = Hs[i], N = H * W;
    const float* in_w   = (const float*)d_in[3 + 8 * i + 0];
    const float* in_b   = (const float*)d_in[3 + 8 * i + 1];
    const float* out_w  = (const float*)d_in[3 + 8 * i + 2];
    const float* out_b  = (const float*)d_in[3 + 8 * i + 3];
    const float* proj_w = (const float*)d_in[3 + 8 * i + 4];
    const float* proj_b = (const float*)d_in[3 + 8 * i + 5];
    const float* down_w = (const float*)d_in[3 + 8 * i + 6];
    const float* down_b = (const float*)d_in[3 + 8 * i + 7];

    // QKV projection: [3C, N] = in_w[3C,C] * cur[C,N] + in_b
    gemm16<<<dim3(N / 64, (3 * C) / 64), 128, 0, stream>>>(
        in_w, cur, in_b, nullptr, qkvb, 3 * C, N, C, 0);

    // windowed attention -> attnb [C, N]
    attn_win<<<dim3(N / 16, Heads[i]), 32, 0, stream>>>(qkvb, attnb, C, H, W);

    // out projection + residual: y = out_w * attnb + out_b + cur
    gemm16<<<dim3(N / 64, C / 64), 128, 0, stream>>>(
        out_w, attnb, out_b, cur, ybuf, C, N, C, 0);

    // 1x1 proj conv + relu -> feats[i] (written straight into d_out)
    gemm16<<<dim3(N / 64, C / 64), 128, 0, stream>>>(
        proj_w, ybuf, proj_b, nullptr, feats[i], C, N, C, 1);

    // 3x3 stride-2 down conv -> next hidden (final stage writes h to out[0])
    float* dst = (i == 2) ? out : hbuf;
    const int total = Dn[i] * (H / 2) * (W / 2);
    conv3x3_s2<<<(total + 255) / 256, 256, 0, stream>>>(
        feats[i], down_w, down_b, dst, C, H, W, Dn[i]);
    cur = dst;
  }
}